// SoftArmGraphNN_70506183131138
// MI455X (gfx1250) — compile-verified
//
#include <hip/hip_runtime.h>
#include <hip/hip_bf16.h>
#include <stdint.h>

typedef __attribute__((ext_vector_type(2))) float v2f;
typedef __attribute__((ext_vector_type(8))) float v8f;

#define LN_EPS 1e-5f

// ---- gfx1250 async global->LDS support (guarded; falls back to plain LDS) --
#if defined(__has_builtin)
#if __has_builtin(__builtin_amdgcn_global_load_async_to_lds_b128)
#define HAVE_ASYNC_LDS 1
#endif
#if __has_builtin(__builtin_amdgcn_s_wait_asynccnt)
#define HAVE_WAIT_ASYNC 1
#endif
#endif

// Builtin prototype (from compiler diagnostic): param0 = v4i in addrspace(1),
// param1 = v4i in addrspace(3).
typedef __attribute__((__vector_size__(4 * sizeof(int)))) int v4i;
typedef __attribute__((address_space(1))) v4i gv4i_t;
typedef __attribute__((address_space(3))) v4i lv4i_t;

static __device__ __forceinline__ void async_wait0() {
#if defined(HAVE_WAIT_ASYNC)
    __builtin_amdgcn_s_wait_asynccnt(0);
#else
    asm volatile("s_wait_asynccnt 0" ::: "memory");
#endif
}

// Stage one 16x64 f32 tile (row stride 68 in LDS) from global, coalesced b128.
// Per-wave private slab: no cross-wave barrier needed; ASYNCcnt is per-wave.
static __device__ __forceinline__ void stage_tile(
    const float* __restrict__ g, long base, int M, float* lds, int lane)
{
#pragma unroll
    for (int j = 0; j < 8; ++j) {
        const int i = lane + 32 * j;      // 0..255 -> 256 float4 slots
        const int row = i >> 4;           // 0..15
        const int c4 = (i & 15) * 4;      // column (float index)
        long gr = base + row;
        if (gr >= M) gr = M - 1;
        const float* gp = g + gr * 64 + c4;
        float* lp = lds + row * 68 + c4;
#if defined(HAVE_ASYNC_LDS)
        __builtin_amdgcn_global_load_async_to_lds_b128(
            (gv4i_t*)(v4i*)gp, (lv4i_t*)(v4i*)lp, 0, 0);
#else
        *(float4*)lp = *(const float4*)gp;
#endif
    }
#if defined(HAVE_ASYNC_LDS)
    async_wait0();
#endif
}

static __device__ __forceinline__ v8f splat8(float f) {
    v8f r;
#pragma unroll
    for (int i = 0; i < 8; ++i) r[i] = f;
    return r;
}

// D = A(16x4 f32) * B(4x16 f32) + C(16x16 f32)  -- full-precision CDNA5 WMMA
static __device__ __forceinline__ v8f wmma_f32(v2f a, v2f b, v8f c) {
    return __builtin_amdgcn_wmma_f32_16x16x4_f32(false, a, false, b, (short)0, c,
                                                 false, false);
}

// ---------------------------------------------------------------------------
// Encoder: x = relu(nf @ W1 + b1) @ W2 + b2      nf:(M,19)  x:(M,64)
// 128 threads = 4 wave32; each wave owns one 16-row tile.
// ---------------------------------------------------------------------------
__global__ void __launch_bounds__(128) encoder_kernel(
    const float* __restrict__ nf, const float* __restrict__ w1,
    const float* __restrict__ b1, const float* __restrict__ w2,
    const float* __restrict__ b2, float* __restrict__ x, int M)
{
    __shared__ float sW1[20 * 64];       // K padded 19 -> 20 (zero row)
    __shared__ float sW2[64 * 64];
    __shared__ float sH[4][16 * 68];     // per-wave h1 staging (padded stride)

    const int tid = threadIdx.x;
    for (int i = tid; i < 20 * 64; i += 128) sW1[i] = (i < 19 * 64) ? w1[i] : 0.0f;
    for (int i = tid; i < 64 * 64; i += 128) sW2[i] = w2[i];
    __syncthreads();

    const int wave = tid >> 5;
    const int lane = tid & 31;
    const int half = lane >> 4;          // A/B fragments: lanes 16-31 hold K+2
    const int l16  = lane & 15;
    const int koff = half * 2;

    const long tile = (long)blockIdx.x * 4 + wave;
    const long base = tile * 16;
    if (base >= M) return;               // wave-uniform: EXEC stays all-ones

    long arow = base + l16;
    if (arow >= M) arow = M - 1;         // clamp (arith select, no divergence)
    const float* nfp = nf + arow * 19;

    v8f acc[4];
#pragma unroll
    for (int t = 0; t < 4; ++t) acc[t] = splat8(b1[t * 16 + l16]);

    // GEMM1: (16x19)@(19x64), K padded to 20
#pragma unroll
    for (int k0 = 0; k0 < 20; k0 += 4) {
        const int ka = k0 + koff;
        v2f a;
        a.x = (ka < 19) ? nfp[ka] : 0.0f;
        a.y = (ka + 1 < 19) ? nfp[ka + 1] : 0.0f;
#pragma unroll
        for (int t = 0; t < 4; ++t) {
            v2f bf;
            bf.x = sW1[ka * 64 + t * 16 + l16];
            bf.y = sW1[(ka + 1) * 64 + t * 16 + l16];
            acc[t] = wmma_f32(a, bf, acc[t]);
        }
    }

    // ReLU, stash h1 (C layout -> row-major LDS), wave-private slab
    float* h = sH[wave];
#pragma unroll
    for (int t = 0; t < 4; ++t)
#pragma unroll
        for (int v = 0; v < 8; ++v) {
            const int r = v + half * 8;
            const float hv = acc[t][v];
            h[r * 68 + t * 16 + l16] = hv > 0.0f ? hv : 0.0f;
        }

    // GEMM2: (16x64)@(64x64); A re-read from LDS in A-fragment layout
#pragma unroll
    for (int t = 0; t < 4; ++t) acc[t] = splat8(b2[t * 16 + l16]);
#pragma unroll
    for (int k0 = 0; k0 < 64; k0 += 4) {
        const int ka = k0 + koff;
        v2f a;
        a.x = h[l16 * 68 + ka];
        a.y = h[l16 * 68 + ka + 1];
#pragma unroll
        for (int t = 0; t < 4; ++t) {
            v2f bf;
            bf.x = sW2[ka * 64 + t * 16 + l16];
            bf.y = sW2[(ka + 1) * 64 + t * 16 + l16];
            acc[t] = wmma_f32(a, bf, acc[t]);
        }
    }

#pragma unroll
    for (int t = 0; t < 4; ++t)
#pragma unroll
        for (int v = 0; v < 8; ++v) {
            const long r = base + v + half * 8;
            if (r < M) x[r * 64 + t * 16 + l16] = acc[t][v];
        }
}

// ---------------------------------------------------------------------------
// Zero agg/cnt (grid-stride)
// ---------------------------------------------------------------------------
__global__ void zero_kernel(float* __restrict__ a, long na,
                            float* __restrict__ c, long nc)
{
    const long i = (long)blockIdx.x * blockDim.x + threadIdx.x;
    const long stride = (long)gridDim.x * blockDim.x;
    for (long j = i; j < na; j += stride) a[j] = 0.0f;
    for (long j = i; j < nc; j += stride) c[j] = 0.0f;
}

// ---------------------------------------------------------------------------
// Edge scatter: one wave32 per edge; 2 feature floats per lane.
// agg[dst] += x[src] ; cnt[dst] += 1
// ---------------------------------------------------------------------------
__global__ void scatter_kernel(const float* __restrict__ x,
                               const long long* __restrict__ ei,  // (B,2,E) i64
                               float* __restrict__ agg, float* __restrict__ cnt,
                               int N, long E, int B)
{
    const long gtid = (long)blockIdx.x * blockDim.x + threadIdx.x;
    const long w = gtid >> 5;
    const int lane = threadIdx.x & 31;
    if (w >= (long)B * E) return;

    const long b = w / E;
    const long e = w - b * E;
    const long long* eb = ei + b * 2 * E;
    const long s = (long)eb[e];
    const long d = (long)eb[E + e];

    const long srow = (b * N + s) * 64;
    const long drow = (b * N + d) * 64;
    const float v0 = x[srow + lane * 2];
    const float v1 = x[srow + lane * 2 + 1];
    atomicAdd(&agg[drow + lane * 2], v0);
    atomicAdd(&agg[drow + lane * 2 + 1], v1);
    if (lane == 0) atomicAdd(&cnt[b * N + d], 1.0f);
}

// ---------------------------------------------------------------------------
// Fused GraphConv + residual + LayerNorm + ReLU (in-place on x)
// y = [x, agg/max(cnt,1)] @ W(128,64) + bias ; x = relu(LN(y + x)*g + b)
// x-tile staged to LDS once via async global->LDS (serves A frags + residual)
// ---------------------------------------------------------------------------
__global__ void __launch_bounds__(128) conv_ln_kernel(
    float* __restrict__ x, const float* __restrict__ agg,
    const float* __restrict__ cnt, const float* __restrict__ W,
    const float* __restrict__ bias, const float* __restrict__ lng,
    const float* __restrict__ lnb, int M)
{
    __shared__ float sW[128 * 64];       // 32 KB
    __shared__ float sX[4][16 * 68];     // 17 KB, per-wave x tiles
    const int tid = threadIdx.x;
    for (int i = tid; i < 128 * 64; i += 128) sW[i] = W[i];
    __syncthreads();

    const int wave = tid >> 5, lane = tid & 31;
    const int half = lane >> 4, l16 = lane & 15, koff = half * 2;
    const long tile = (long)blockIdx.x * 4 + wave;
    const long base = tile * 16;
    if (base >= M) return;

    // async-stage this wave's 16x64 x tile into LDS (coalesced b128)
    float* xt = sX[wave];
    stage_tile(x, base, M, xt, lane);

    long arow = base + l16;
    if (arow >= M) arow = M - 1;
    const float* ar = agg + arow * 64;
    const float rcp = 1.0f / fmaxf(cnt[arow], 1.0f);

    v8f acc[4];
#pragma unroll
    for (int t = 0; t < 4; ++t) acc[t] = splat8(bias[t * 16 + l16]);

#pragma unroll
    for (int k0 = 0; k0 < 128; k0 += 4) {
        const int ka = k0 + koff;
        v2f a;
        if (k0 < 64) {                       // uniform per k0
            a.x = xt[l16 * 68 + ka];
            a.y = xt[l16 * 68 + ka + 1];
        } else {
            a.x = ar[ka - 64] * rcp;
            a.y = ar[ka - 63] * rcp;
        }
#pragma unroll
        for (int t = 0; t < 4; ++t) {
            v2f bf;
            bf.x = sW[ka * 64 + t * 16 + l16];
            bf.y = sW[(ka + 1) * 64 + t * 16 + l16];
            acc[t] = wmma_f32(a, bf, acc[t]);
        }
    }

    // residual add (from LDS tile) + row statistics. C layout:
    // row = base + v + 8*half; 64 features of a row live in 4 n-tiles
    // across the 16 lanes of a half.
    float sum[8], sq[8];
#pragma unroll
    for (int v = 0; v < 8; ++v) { sum[v] = 0.0f; sq[v] = 0.0f; }
#pragma unroll
    for (int v = 0; v < 8; ++v) {
        const int lr = v + half * 8;
#pragma unroll
        for (int t = 0; t < 4; ++t) {
            const float val = acc[t][v] + xt[lr * 68 + t * 16 + l16];
            acc[t][v] = val;
            sum[v] += val;
            sq[v]  += val * val;
        }
    }
    // reduce across the 16 lanes of each half (masks < 16 never cross halves)
#pragma unroll
    for (int v = 0; v < 8; ++v)
#pragma unroll
        for (int m = 1; m < 16; m <<= 1) {
            sum[v] += __shfl_xor(sum[v], m, 32);
            sq[v]  += __shfl_xor(sq[v], m, 32);
        }

    float gv[4], bv[4];
#pragma unroll
    for (int t = 0; t < 4; ++t) { gv[t] = lng[t * 16 + l16]; bv[t] = lnb[t * 16 + l16]; }

#pragma unroll
    for (int v = 0; v < 8; ++v) {
        const float mu  = sum[v] * (1.0f / 64.0f);
        const float var = sq[v] * (1.0f / 64.0f) - mu * mu;
        const float rs  = rsqrtf(var + LN_EPS);
        const long r = base + v + half * 8;
        if (r < M) {
#pragma unroll
            for (int t = 0; t < 4; ++t) {
                const float o = (acc[t][v] - mu) * rs * gv[t] + bv[t];
                x[r * 64 + t * 16 + l16] = o > 0.0f ? o : 0.0f;
            }
        }
    }
}

// ---------------------------------------------------------------------------
// Head: out = x @ Wo(64,32) + bo   (x tile async-staged to LDS)
// ---------------------------------------------------------------------------
__global__ void __launch_bounds__(128) out_kernel(
    const float* __restrict__ x, const float* __restrict__ W,
    const float* __restrict__ b, float* __restrict__ out, int M)
{
    __shared__ float sW[64 * 32];
    __shared__ float sX[4][16 * 68];
    const int tid = threadIdx.x;
    for (int i = tid; i < 64 * 32; i += 128) sW[i] = W[i];
    __syncthreads();

    const int wave = tid >> 5, lane = tid & 31;
    const int half = lane >> 4, l16 = lane & 15, koff = half * 2;
    const long tile = (long)blockIdx.x * 4 + wave;
    const long base = tile * 16;
    if (base >= M) return;

    float* xt = sX[wave];
    stage_tile(x, base, M, xt, lane);

    v8f acc[2];
#pragma unroll
    for (int t = 0; t < 2; ++t) acc[t] = splat8(b[t * 16 + l16]);
#pragma unroll
    for (int k0 = 0; k0 < 64; k0 += 4) {
        const int ka = k0 + koff;
        v2f a;
        a.x = xt[l16 * 68 + ka];
        a.y = xt[l16 * 68 + ka + 1];
#pragma unroll
        for (int t = 0; t < 2; ++t) {
            v2f bf;
            bf.x = sW[ka * 32 + t * 16 + l16];
            bf.y = sW[(ka + 1) * 32 + t * 16 + l16];
            acc[t] = wmma_f32(a, bf, acc[t]);
        }
    }
#pragma unroll
    for (int t = 0; t < 2; ++t)
#pragma unroll
        for (int v = 0; v < 8; ++v) {
            const long r = base + v + half * 8;
            if (r < M) out[r * 32 + t * 16 + l16] = acc[t][v];
        }
}

// ---------------------------------------------------------------------------
extern "C" void kernel_launch(void* const* d_in, const int* in_sizes, int n_in,
                              void* d_out, int out_size, void* d_ws, size_t ws_size,
                              hipStream_t stream)
{
    const float*     nf     = (const float*)d_in[0];
    const long long* ei     = (const long long*)d_in[1];   // int64 edge indices
    const float*     enc_w1 = (const float*)d_in[3];
    const float*     enc_b1 = (const float*)d_in[4];
    const float*     enc_w2 = (const float*)d_in[5];
    const float*     enc_b2 = (const float*)d_in[6];
    const float*     conv_w = (const float*)d_in[7];       // (3,128,64)
    const float*     conv_b = (const float*)d_in[8];       // (3,64)
    const float*     ln_g   = (const float*)d_in[9];       // (3,64)
    const float*     ln_b   = (const float*)d_in[10];      // (3,64)
    const float*     out_w  = (const float*)d_in[11];      // (64,32)
    const float*     out_b  = (const float*)d_in[12];      // (32,)

    const int  Bn = 16, IN = 19, H = 64, L = 3;
    const long N = (long)in_sizes[0] / ((long)Bn * IN);
    const long E = (long)in_sizes[1] / ((long)Bn * 2);
    const long M = (long)Bn * N;

    float* x   = (float*)d_ws;        // (M,64)
    float* agg = x + M * H;           // (M,64)
    float* cnt = agg + M * H;         // (M,)

    const long tiles = (M + 15) / 16;
    const int  gemmBlocks = (int)((tiles + 3) / 4);

    encoder_kernel<<<gemmBlocks, 128, 0, stream>>>(nf, enc_w1, enc_b1, enc_w2,
                                                   enc_b2, x, (int)M);

    const long warps = (long)Bn * E;
    const int  sblocks = (int)((warps * 32 + 255) / 256);

    for (int l = 0; l < L; ++l) {
        zero_kernel<<<4096, 256, 0, stream>>>(agg, M * H, cnt, M);
        scatter_kernel<<<sblocks, 256, 0, stream>>>(x, ei, agg, cnt, (int)N, E, Bn);
        conv_ln_kernel<<<gemmBlocks, 128, 0, stream>>>(
            x, agg, cnt, conv_w + (long)l * 128 * 64, conv_b + l * 64,
            ln_g + l * 64, ln_b + l * 64, (int)M);
    }

    out_kernel<<<gemmBlocks, 128, 0, stream>>>(x, out_w, out_b, (float*)d_out, (int)M);
}